// NavierStokes2DSpectral_19104014532851
// MI455X (gfx1250) — compile-verified
//
#include <hip/hip_runtime.h>
#include <math.h>

// ---------------------------------------------------------------------------
// 2D Navier-Stokes pseudo-spectral step for MI455X (gfx1250).
// All FFTs are 1024-pt four-step (32x32) with the 32-pt DFTs done as complex
// 32x32x32 matmuls on V_WMMA_F32_16X16X4_F32 (full fp32 precision).
// Planar complex storage; spectral arrays kept TRANSPOSED:
//   S[b][ky(0..512)][kx(0..1023)]  so the 1024-mode axis is contiguous.
// Pointwise physics (spectral multipliers, advection MAC, 2/3-filter and
// CN/RK stage update) are fused into the FFT kernel's load/store phases.
// Pure global->LDS copies use the CDNA5 async data movers (ASYNCcnt).
// ---------------------------------------------------------------------------

#define PI_F 3.14159265358979323846f
#define NB 8
#define NN 1024
#define NH 513
#define VISC 1e-3f
#define DRAG 0.1f
#define DTC  1e-3f

typedef __attribute__((ext_vector_type(2))) float v2f;
typedef __attribute__((ext_vector_type(8))) float v8f;

__device__ __forceinline__ v8f wmma4(v2f a, v2f b, v8f c) {
  // D(16x16,f32) = A(16x4,f32) * B(4x16,f32) + C
  return __builtin_amdgcn_wmma_f32_16x16x4_f32(false, a, false, b, (short)0, c,
                                               false, false);
}

// --- CDNA5 async global->LDS movers (tracked by ASYNCcnt) ------------------
// LDS offset = low 32 bits of the generic shared-aperture address.
__device__ __forceinline__ void async_ld_b128(const float* g, float* lds) {
  unsigned l = (unsigned)(unsigned long long)lds;
  asm volatile("global_load_async_to_lds_b128 %0, %1, off"
               :
               : "v"(l), "v"(g)
               : "memory");
}
__device__ __forceinline__ void async_ld_b32(const float* g, float* lds) {
  unsigned l = (unsigned)(unsigned long long)lds;
  asm volatile("global_load_async_to_lds_b32 %0, %1, off"
               :
               : "v"(l), "v"(g)
               : "memory");
}
__device__ __forceinline__ void async_wait0() {
  asm volatile("s_wait_asynccnt 0x0" ::: "memory");
}

// A-fragment (16x4) per ISA layout: VGPR0 lanes0-15 K=0, lanes16-31 K=2;
// VGPR1 K=1 / K=3.  M = lane&15 (+ 16*mt).
__device__ __forceinline__ v2f ldAfrag(const float* A, int mt, int kb, int lane) {
  int M = mt * 16 + (lane & 15);
  int K = kb * 4 + ((lane >> 4) << 1);
  v2f r;
  r.x = A[M * 32 + K];
  r.y = A[M * 32 + K + 1];
  return r;
}
// B-fragment (4x16), mirror layout: VGPR0 lanes0-15 K=0, lanes16-31 K=2.
__device__ __forceinline__ v2f ldBfrag(const float* Bm, int kb, int nt, int lane) {
  int N = nt * 16 + (lane & 15);
  int K = kb * 4 + ((lane >> 4) << 1);
  v2f r;
  r.x = Bm[K * 32 + N];
  r.y = Bm[(K + 1) * 32 + N];
  return r;
}

// Complex 32x32x32 matmul D = L * R on LDS buffers (row-major, stride 32 for
// inputs, dstride for output).  One wave per call.  Optional fused twiddle
// D[M][N] *= exp(-i * twsign * 2*pi*M*N/1024) applied at store.
__device__ __forceinline__ void cmatmul32(float* dRe, float* dIm, int dstride,
                                          const float* lRe, const float* lIm,
                                          const float* rRe, const float* rIm,
                                          int lane, float twsign) {
#pragma unroll
  for (int mt = 0; mt < 2; ++mt) {
    v2f ar[8], ai[8], an[8];
#pragma unroll
    for (int kb = 0; kb < 8; ++kb) {
      ar[kb] = ldAfrag(lRe, mt, kb, lane);
      ai[kb] = ldAfrag(lIm, mt, kb, lane);
      an[kb].x = -ai[kb].x;
      an[kb].y = -ai[kb].y;
    }
#pragma unroll
    for (int nt = 0; nt < 2; ++nt) {
      v8f cr = {0.f, 0.f, 0.f, 0.f, 0.f, 0.f, 0.f, 0.f};
      v8f ci = {0.f, 0.f, 0.f, 0.f, 0.f, 0.f, 0.f, 0.f};
#pragma unroll
      for (int kb = 0; kb < 8; ++kb) {
        v2f br = ldBfrag(rRe, kb, nt, lane);
        v2f bi = ldBfrag(rIm, kb, nt, lane);
        cr = wmma4(ar[kb], br, cr);   // + Lr*Rr
        cr = wmma4(an[kb], bi, cr);   // - Li*Ri
        ci = wmma4(ar[kb], bi, ci);   // + Lr*Ri
        ci = wmma4(ai[kb], br, ci);   // + Li*Rr
      }
      int M0 = mt * 16 + ((lane >> 4) << 3);
      int N = nt * 16 + (lane & 15);
#pragma unroll
      for (int r = 0; r < 8; ++r) {
        int M = M0 + r;
        float re = cr[r], im = ci[r];
        if (twsign != 0.f) {
          float ang = twsign * (-2.0f * PI_F / 1024.0f) * (float)(M * N);
          float s, c;
          __sincosf(ang, &s, &c);
          float tre = re * c - im * s;
          float tim = re * s + im * c;
          re = tre;
          im = tim;
        }
        dRe[M * dstride + N] = re;
        dIm[M * dstride + N] = im;
      }
    }
  }
}

// Batched 1024-pt FFT along contiguous rows.  4 waves/WG, one row per wave.
// mode: 0 = C2C, 1 = R2C (real in, store k=0..512), 2 = C2R (Hermitian 513 in,
// store 1024 reals).  dirsign: +1 forward (e^{-i}), -1 inverse (e^{+i}).
// Fusions:
//  do_mult (mode 0 load):  x = i*fac(kx,ky)*u,  fac = sgn*2pi*K*(psi?1/lap:1)
//  macmode (mode 2 store): 1: acc = pair*val ; 2: acc = -(acc + pair*val)
//  do_stage (mode 0 store): h = filt*F + beta*h ; u = ((1+muL)u + gdt*h)/(1-muL)
__global__ __launch_bounds__(128) void k_fft_rows(
    const float* __restrict__ inRe, const float* __restrict__ inIm,
    float* __restrict__ outRe, float* __restrict__ outIm, int nrows,
    int inStride, int outStride, int mode, float dirsign, float scale,
    int do_mult, int use_kx, int use_psi, float sgn,
    int macmode, const float* __restrict__ pair, float* __restrict__ accp,
    int do_stage, float* __restrict__ hRe, float* __restrict__ hIm,
    float* __restrict__ uRe, float* __restrict__ uIm, float beta, float gdt,
    float mu) {
  __shared__ float dftRe[1024];
  __shared__ float dftIm[1024];
  __shared__ float Are[4][1056];  // row buffer / final Z (stride 33)
  __shared__ float Aim[4][1056];
  __shared__ float Bre[4][1024];  // intermediate Y (stride 32)
  __shared__ float Bim[4][1024];

  int tid = threadIdx.x;
  int lane = tid & 31;
  int w = tid >> 5;

  // DFT32 matrix: W[m][k] = exp(-i*dirsign*2*pi*m*k/32)
  for (int e = tid; e < 1024; e += 128) {
    int m = e >> 5, k = e & 31;
    float ang = -dirsign * (2.0f * PI_F / 32.0f) * (float)(m * k);
    float s, c;
    __sincosf(ang, &s, &c);
    dftRe[e] = c;
    dftIm[e] = s;
  }
  __syncthreads();

  long long row = (long long)blockIdx.x * 4 + w;  // nrows is a multiple of 4
  float* are = Are[w];
  float* aim = Aim[w];
  float* bre = Bre[w];
  float* bim = Bim[w];
  const float* ire = inRe + row * inStride;

  if (mode == 0) {
    const float* iim = inIm + row * inStride;
    if (do_mult) {
      int j = (int)(row % NH);  // ky index
      float ky = (float)j;
      for (int e = lane; e < 1024; e += 32) {
        float kx = (e < 512) ? (float)e : (float)(e - 1024);
        float K = use_kx ? kx : ky;
        float fac = sgn * 2.0f * PI_F * K;
        if (use_psi) {
          float lap = -(4.0f * PI_F * PI_F) * (kx * kx + ky * ky);
          if (e == 0 && j == 0) lap = 1.0f;
          fac = fac / lap;
        }
        float ur = ire[e], ui = iim[e];
        are[e] = -fac * ui;
        aim[e] = fac * ur;
      }
    } else {
      // Pure copy: async global->LDS, 16B per lane per issue.
      for (int e = lane * 4; e < 1024; e += 128) {
        async_ld_b128(ire + e, are + e);
        async_ld_b128(iim + e, aim + e);
      }
      async_wait0();
    }
  } else if (mode == 1) {
    for (int e = lane * 4; e < 1024; e += 128)
      async_ld_b128(ire + e, are + e);
    for (int e = lane; e < 1024; e += 32) aim[e] = 0.f;
    async_wait0();
  } else {
    const float* iim = inIm + row * inStride;
    for (int e = lane; e < 513; e += 32) {
      float vr = ire[e], vi = iim[e];
      are[e] = vr;
      aim[e] = vi;
      if (e > 0 && e < 512) {  // Hermitian mirror
        are[1024 - e] = vr;
        aim[1024 - e] = -vi;
      }
    }
  }
  __syncthreads();

  // Step 1: Y[k1][n2] = sum_n1 W32[k1][n1] * x[32*n1+n2], fused twiddle.
  cmatmul32(bre, bim, 32, dftRe, dftIm, are, aim, lane, dirsign);
  __syncthreads();
  // Step 2: Z[k1][k2] = sum_n2 Y[k1][n2] * W32[n2][k2] (stride-33 output).
  cmatmul32(are, aim, 33, bre, bim, dftRe, dftIm, lane, 0.f);
  __syncthreads();

  // out[k1 + 32*k2] = Z[k1][k2]
  if (mode == 2) {
    long long ob = row * outStride;
    for (int k = lane; k < 1024; k += 32) {
      float val = are[(k & 31) * 33 + (k >> 5)] * scale;
      if (macmode == 0) {
        outRe[ob + k] = val;
      } else {
        float p = pair[ob + k] * val;
        if (macmode == 1)
          accp[ob + k] = p;
        else
          accp[ob + k] = -(accp[ob + k] + p);
      }
    }
  } else if (mode == 0 && do_stage) {
    int j = (int)(row % NH);  // ky index
    float ky = (float)j;
    long long ob = row * 1024;
    for (int k = lane; k < 1024; k += 32) {
      float fr = are[(k & 31) * 33 + (k >> 5)];
      float fi = aim[(k & 31) * 33 + (k >> 5)];
      // 2/3-rule filter, Python-exact: i<341 or i>=683, j<341
      float fl = (((k < 341) || (k >= 683)) && (j < 341)) ? 1.0f : 0.0f;
      float hr = fr * fl + beta * hRe[ob + k];
      float hi = fi * fl + beta * hIm[ob + k];
      hRe[ob + k] = hr;
      hIm[ob + k] = hi;
      float kx = (k < 512) ? (float)k : (float)(k - 1024);
      float lap = -(4.0f * PI_F * PI_F) * (kx * kx + ky * ky);
      float L = VISC * lap - DRAG;
      float inv = 1.0f / (1.0f - mu * L);
      float cu = (1.0f + mu * L) * inv;
      float ch = gdt * inv;
      uRe[ob + k] = cu * uRe[ob + k] + ch * hr;
      uIm[ob + k] = cu * uIm[ob + k] + ch * hi;
    }
  } else {
    long long ob = row * outStride;
    float* oim = outIm + row * outStride;
    int olen = (mode == 1) ? 513 : 1024;
    for (int k = lane; k < olen; k += 32) {
      outRe[ob + k] = are[(k & 31) * 33 + (k >> 5)] * scale;
      oim[k] = aim[(k & 31) * 33 + (k >> 5)] * scale;
    }
  }
}

// Batched planar-complex tiled transpose: (B,R,C) -> (B,C,R)
__global__ __launch_bounds__(256) void k_transpose(
    const float* __restrict__ iRe, const float* __restrict__ iIm,
    float* __restrict__ oRe, float* __restrict__ oIm, int R, int C) {
  __shared__ float tre[32][33];
  __shared__ float tim[32][33];
  int b = blockIdx.z;
  long long base = (long long)b * R * C;
  int c0 = blockIdx.x * 32;
  int r0 = blockIdx.y * 32;
  int tx = threadIdx.x & 31;
  int ty = threadIdx.x >> 5;  // 0..7
  bool full = (r0 + 32 <= R) && (c0 + 32 <= C);
  if (full) {
    // Interior tile: async global->LDS, one b32 per lane per row.
    for (int i = ty; i < 32; i += 8) {
      const float* gre = iRe + base + (long long)(r0 + i) * C + (c0 + tx);
      const float* gim = iIm + base + (long long)(r0 + i) * C + (c0 + tx);
      async_ld_b32(gre, &tre[i][tx]);
      async_ld_b32(gim, &tim[i][tx]);
    }
    async_wait0();
  } else {
    for (int i = ty; i < 32; i += 8) {
      int r = r0 + i, c = c0 + tx;
      if (r < R && c < C) {
        tre[i][tx] = iRe[base + (long long)r * C + c];
        tim[i][tx] = iIm[base + (long long)r * C + c];
      }
    }
  }
  __syncthreads();
  for (int i = ty; i < 32; i += 8) {
    int c = c0 + i, r = r0 + tx;
    if (r < R && c < C) {
      oRe[base + (long long)c * R + r] = tre[tx][i];
      oIm[base + (long long)c * R + r] = tim[tx][i];
    }
  }
}

// Pack outputs: 4 planes of (B,1024,513) in reference orientation from the
// transposed spectral layout.  per = B*1024*513.
__global__ __launch_bounds__(256) void k_pack(
    const float* __restrict__ uRe, const float* __restrict__ uIm,
    const float* __restrict__ v0Re, const float* __restrict__ v0Im,
    float* __restrict__ out, long long per) {
  long long idx = (long long)blockIdx.x * blockDim.x + threadIdx.x;
  if (idx >= per) return;
  int j = (int)(idx % 513);
  long long t = idx / 513;
  int i = (int)(t % 1024);
  int b = (int)(t / 1024);
  long long s = ((long long)b * 513 + j) * 1024 + i;
  float ur = uRe[s], ui = uIm[s];
  out[idx] = ur;
  out[per + idx] = ui;
  out[2 * per + idx] = (ur - v0Re[s]) * (1.0f / DTC);
  out[3 * per + idx] = (ui - v0Im[s]) * (1.0f / DTC);
}

// ---------------------------------------------------------------------------

extern "C" void kernel_launch(void* const* d_in, const int* in_sizes, int n_in,
                              void* d_out, int out_size, void* d_ws,
                              size_t ws_size, hipStream_t stream) {
  (void)in_sizes; (void)n_in; (void)out_size; (void)ws_size;
  const float* w0 = (const float*)d_in[0];
  float* out = (float*)d_out;

  const long long SZS = (long long)NB * NH * NN;  // spectral planar field
  const long long SZR = (long long)NB * NN * NN;  // real field

  float* p = (float*)d_ws;
  float* v0re = p; p += SZS;  float* v0im = p; p += SZS;
  float* ure  = p; p += SZS;  float* uim  = p; p += SZS;
  float* hre  = p; p += SZS;  float* him  = p; p += SZS;
  float* t1re = p; p += SZS;  float* t1im = p; p += SZS;
  float* t2re = p; p += SZS;  float* t2im = p; p += SZS;
  float* racc = p; p += SZR;
  float* r0   = p; p += SZR;

  const long long PW = 256;
  dim3 pw_s((unsigned)((SZS + PW - 1) / PW));
  dim3 tr_f(17, 32, NB);  // (B,1024,513) -> (B,513,1024)
  dim3 tr_b(32, 17, NB);  // (B,513,1024) -> (B,1024,513)
  const int ROWS_Y = NB * NN / 4;  // 2048 blocks, rows over y-axis
  const int ROWS_X = NB * NH / 4;  // 1026 blocks, rows over x-axis

  // Inverse of (i*fac)*u -> real field with optional fused advection MAC.
  auto inv_field = [&](int use_kx, int use_psi, float sgn, float* realdst,
                       int macmode) {
    // inverse C2C over kx, spectral multiplier fused into the load
    k_fft_rows<<<ROWS_X, 128, 0, stream>>>(
        ure, uim, t2re, t2im, NB * NH, NN, NN, 0, -1.f, 1.0f / 1024.0f,
        1, use_kx, use_psi, sgn, 0, nullptr, nullptr, 0, nullptr, nullptr,
        nullptr, nullptr, 0.f, 0.f, 0.f);
    k_transpose<<<tr_b, 256, 0, stream>>>(t2re, t2im, t1re, t1im, NH, NN);
    // inverse C2R over ky, optional fused MAC into racc
    k_fft_rows<<<ROWS_Y, 128, 0, stream>>>(
        t1re, t1im, realdst, nullptr, NB * NN, NH, NN, 2, -1.f, 1.0f / 1024.0f,
        0, 0, 0, 0.f, macmode, r0, racc, 0, nullptr, nullptr, nullptr, nullptr,
        0.f, 0.f, 0.f);
  };

  // vort_hat = rfft2(w0) -> u ; v0 = copy ; h = 0
  k_fft_rows<<<ROWS_Y, 128, 0, stream>>>(
      w0, nullptr, t1re, t1im, NB * NN, NN, NH, 1, 1.f, 1.f, 0, 0, 0, 0.f, 0,
      nullptr, nullptr, 0, nullptr, nullptr, nullptr, nullptr, 0.f, 0.f, 0.f);
  k_transpose<<<tr_f, 256, 0, stream>>>(t1re, t1im, t2re, t2im, NN, NH);
  k_fft_rows<<<ROWS_X, 128, 0, stream>>>(
      t2re, t2im, ure, uim, NB * NH, NN, NN, 0, 1.f, 1.f, 0, 0, 0, 0.f, 0,
      nullptr, nullptr, 0, nullptr, nullptr, nullptr, nullptr, 0.f, 0.f, 0.f);
  hipMemcpyAsync(v0re, ure, SZS * sizeof(float), hipMemcpyDeviceToDevice, stream);
  hipMemcpyAsync(v0im, uim, SZS * sizeof(float), hipMemcpyDeviceToDevice, stream);
  hipMemsetAsync(hre, 0, SZS * sizeof(float), stream);
  hipMemsetAsync(him, 0, SZS * sizeof(float), stream);

  const float ALPHAS[6] = {0.0f, 0.1496590219993f, 0.3704009573644f,
                           0.6222557631345f, 0.9582821306748f, 1.0f};
  const float BETAS[5] = {0.0f, -0.4178904745f, -1.192151694643f,
                          -1.697784692471f, -1.514183444257f};
  const float GAMMAS[5] = {0.1496590219993f, 0.3792103129999f,
                           0.8229550293869f, 0.6994504559488f,
                           0.1530572479681f};

  for (int k = 0; k < 5; ++k) {
    // vx = irfft2(2pi*i*ky*psi), psi=-u/lap -> fac = -2pi*ky/lap
    inv_field(0, 1, -1.f, r0, 0);
    // gx = irfft2(2pi*i*kx*u);  acc = vx*gx
    inv_field(1, 0, 1.f, nullptr, 1);
    // vy = irfft2(-2pi*i*kx*psi) -> fac = +2pi*kx/lap
    inv_field(1, 1, 1.f, r0, 0);
    // gy = irfft2(2pi*i*ky*u);  acc = -(acc + vy*gy)  == advection
    inv_field(0, 0, 1.f, nullptr, 2);

    // F = rfft2(advection); filter + h,u update fused into the final C2C
    float mu = 0.5f * DTC * (ALPHAS[k + 1] - ALPHAS[k]);
    k_fft_rows<<<ROWS_Y, 128, 0, stream>>>(
        racc, nullptr, t1re, t1im, NB * NN, NN, NH, 1, 1.f, 1.f, 0, 0, 0, 0.f,
        0, nullptr, nullptr, 0, nullptr, nullptr, nullptr, nullptr, 0.f, 0.f,
        0.f);
    k_transpose<<<tr_f, 256, 0, stream>>>(t1re, t1im, t2re, t2im, NN, NH);
    k_fft_rows<<<ROWS_X, 128, 0, stream>>>(
        t2re, t2im, t1re, t1im, NB * NH, NN, NN, 0, 1.f, 1.f, 0, 0, 0, 0.f, 0,
        nullptr, nullptr, 1, hre, him, ure, uim, BETAS[k], GAMMAS[k] * DTC, mu);
  }

  k_pack<<<pw_s, PW, 0, stream>>>(ure, uim, v0re, v0im, out, SZS);
}